// Operator_62414464745996
// MI455X (gfx1250) — compile-verified
//
#include <hip/hip_runtime.h>

// ---- problem constants (from reference) ----
#define B_    8
#define T_    12
#define C_    5
#define H_    64
#define W_    128
#define PAD_  2
#define KERN_ 4
#define HP_   (H_ + 2 * PAD_)   // 68
#define WP_   (W_ + 2 * PAD_)   // 132
#define NY_   33                // len(arange(0, 66, 2))
#define NX_   65                // len(arange(0, 130, 2))
#define F_    (C_ * KERN_ * KERN_)  // 80 features per snapshot row

typedef __attribute__((ext_vector_type(2))) float v2f;
typedef __attribute__((ext_vector_type(8))) float v8f;

__global__ void pod_zero_kernel(float* __restrict__ out, int n) {
    int i = blockIdx.x * blockDim.x + threadIdx.x;
    if (i < n) out[i] = 0.0f;
}

// One wave (32 lanes) per 4x4 window. P is the T x F snapshot matrix of the
// window (T=12 padded to 16, F=80).  recon = W @ P with W = V' diag(w) V,
// w_m = (e_m > 1) ? 1 : (e_m/(e_m+1))^2, from the eigen-decomposition of the
// 12x12 Gram G = P P'.  Both GEMMs run on V_WMMA_F32_16X16X4_F32.
__global__ __launch_bounds__(32) void pod_window_kernel(const float* __restrict__ x,
                                                        float* __restrict__ out) {
    __shared__ float P[16 * F_];     // 16 x 80, rows 12..15 zeroed
    __shared__ float Gs[16 * 16];    // Gram (then Jacobi-diagonalized in place)
    __shared__ float Vs[12 * 12];    // accumulated eigenvectors (columns)
    __shared__ float Ws[16 * 16];    // spectral filter, zero outside 12x12

    const int lane = threadIdx.x;          // 0..31
    const int wid  = blockIdx.x;           // window id
    const int b    = wid / (NY_ * NX_);
    const int rem  = wid % (NY_ * NX_);
    const int iy   = rem / NX_;
    const int ix   = rem % NX_;
    const int ph   = 2 * iy;               // window origin in padded coords
    const int pw   = 2 * ix;

    // ---- cooperative load: 12*80 = 960 elements, 30 per lane ----
    for (int i = 0; i < 30; ++i) {
        int e  = lane + 32 * i;
        int t  = e / F_;
        int f  = e % F_;
        int c  = f >> 4;
        int kh = (f >> 2) & 3;
        int kw = f & 3;
        int rr = ph + kh - PAD_;                    // reflect pad on H
        rr = rr < 0 ? -rr : rr;
        rr = rr > (H_ - 1) ? 2 * (H_ - 1) - rr : rr;
        int cc = (pw + kw - PAD_ + W_) % W_;        // wrap pad on W
        P[t * F_ + f] = x[(((size_t)(b * T_ + t) * C_ + c) * H_ + rr) * W_ + cc];
    }
    for (int i = 0; i < 10; ++i)                    // zero rows 12..15
        P[12 * F_ + lane + 32 * i] = 0.0f;
    __syncthreads();

    const int m16 = lane & 15;
    const int hi  = lane >> 4;                      // 0: K pair {0,1}; 1: K pair {2,3}

    // ---- Gram = P P^T via f32 WMMA, K=80 in 20 chunks of 4 ----
    // For A (16x4) and B=P^T (4x16) the per-lane register image is identical.
    v8f g = {};
    for (int k0 = 0; k0 < F_; k0 += 4) {
        int idx = m16 * F_ + k0 + 2 * hi;
        v2f a;
        a.x = P[idx];
        a.y = P[idx + 1];
        g = __builtin_amdgcn_wmma_f32_16x16x4_f32(false, a, false, a,
                                                  (short)0, g, false, false);
    }
    // D layout: lane holds G[m, n], n = lane%16, m = v + 8*(lane/16)
#pragma unroll
    for (int v = 0; v < 8; ++v)
        Gs[(v + 8 * hi) * 16 + m16] = g[v];
    __syncthreads();

    // ---- lane-0 serial cyclic Jacobi on the 12x12 Gram (in LDS) ----
    if (lane == 0) {
        for (int i = 0; i < 12; ++i)
            for (int j = 0; j < 12; ++j)
                Vs[i * 12 + j] = (i == j) ? 1.0f : 0.0f;
        for (int sweep = 0; sweep < 12; ++sweep) {
            for (int p = 0; p < 11; ++p) {
                for (int q = p + 1; q < 12; ++q) {
                    float apq = Gs[p * 16 + q];
                    if (fabsf(apq) < 1e-12f) continue;
                    float app = Gs[p * 16 + p];
                    float aqq = Gs[q * 16 + q];
                    float tau = (aqq - app) / (2.0f * apq);
                    float t   = (tau >= 0.0f ? 1.0f : -1.0f) /
                                (fabsf(tau) + sqrtf(1.0f + tau * tau));
                    float c   = rsqrtf(1.0f + t * t);
                    float s   = t * c;
                    for (int k = 0; k < 12; ++k) {          // G <- G J
                        float gkp = Gs[k * 16 + p];
                        float gkq = Gs[k * 16 + q];
                        Gs[k * 16 + p] = c * gkp - s * gkq;
                        Gs[k * 16 + q] = s * gkp + c * gkq;
                    }
                    for (int k = 0; k < 12; ++k) {          // G <- J^T G
                        float gpk = Gs[p * 16 + k];
                        float gqk = Gs[q * 16 + k];
                        Gs[p * 16 + k] = c * gpk - s * gqk;
                        Gs[q * 16 + k] = s * gpk + c * gqk;
                    }
                    for (int k = 0; k < 12; ++k) {          // V <- V J
                        float vkp = Vs[k * 12 + p];
                        float vkq = Vs[k * 12 + q];
                        Vs[k * 12 + p] = c * vkp - s * vkq;
                        Vs[k * 12 + q] = s * vkp + c * vkq;
                    }
                }
            }
        }
    }
    __syncthreads();

    // ---- W = V diag(w) V^T, built by all lanes (256 entries, 8 per lane) ----
    for (int i = 0; i < 8; ++i) {
        int e  = lane + 32 * i;      // 0..255
        int ra = e >> 4;
        int rb = e & 15;
        float acc = 0.0f;
        if (ra < 12 && rb < 12) {
            for (int m = 0; m < 12; ++m) {
                float ev = Gs[m * 16 + m];
                ev = ev < 0.0f ? 0.0f : ev;
                float frac = ev / (ev + 1.0f);
                float wgt  = (ev > 1.0f) ? 1.0f : frac * frac;
                acc += wgt * Vs[ra * 12 + m] * Vs[rb * 12 + m];
            }
        }
        Ws[e] = acc;
    }
    __syncthreads();

    // ---- recon = W @ P via f32 WMMA: 5 column tiles x 4 K-chunks ----
    for (int n0 = 0; n0 < F_; n0 += 16) {
        v8f r = {};
#pragma unroll
        for (int k0 = 0; k0 < 16; k0 += 4) {
            int koff = k0 + 2 * hi;
            v2f a;                                   // A = W[:, k0:k0+4]
            a.x = Ws[m16 * 16 + koff];
            a.y = Ws[m16 * 16 + koff + 1];
            v2f bb;                                  // B = P[k0:k0+4, n0:n0+16]
            bb.x = P[koff * F_ + n0 + m16];
            bb.y = P[(koff + 1) * F_ + n0 + m16];
            r = __builtin_amdgcn_wmma_f32_16x16x4_f32(false, a, false, bb,
                                                      (short)0, r, false, false);
        }
        // scatter-add: lane holds recon[t, f], f = n0 + lane%16, t = v + 8*hi
        int f  = n0 + m16;
        int c  = f >> 4;
        int kh = (f >> 2) & 3;
        int kw = f & 3;
        size_t hwoff = (size_t)(ph + kh) * WP_ + (pw + kw);
#pragma unroll
        for (int v = 0; v < 8; ++v) {
            int t = v + 8 * hi;
            if (t < 12) {
                atomicAdd(&out[((size_t)(b * T_ + t) * C_ + c) * (HP_ * WP_) + hwoff],
                          r[v]);
            }
        }
    }
}

// Divide by the analytically-known overlap count: a padded cell (r,c) is
// covered by 2 window-rows unless r<2 or r>=66, same for columns at 2/130.
__global__ void pod_norm_kernel(float* __restrict__ out, int n) {
    int i = blockIdx.x * blockDim.x + threadIdx.x;
    if (i >= n) return;
    int hw = i % (HP_ * WP_);
    int r  = hw / WP_;
    int c  = hw % WP_;
    float nh = (r < 2 || r >= HP_ - 2) ? 1.0f : 2.0f;
    float nw = (c < 2 || c >= WP_ - 2) ? 1.0f : 2.0f;
    out[i] *= 1.0f / (nh * nw);
}

extern "C" void kernel_launch(void* const* d_in, const int* in_sizes, int n_in,
                              void* d_out, int out_size, void* d_ws, size_t ws_size,
                              hipStream_t stream) {
    const float* x = (const float*)d_in[0];
    float* out = (float*)d_out;
    const int n = out_size;                       // 8*12*5*68*132 = 4,308,480

    pod_zero_kernel<<<(n + 255) / 256, 256, 0, stream>>>(out, n);
    pod_window_kernel<<<B_ * NY_ * NX_, 32, 0, stream>>>(x, out);
    pod_norm_kernel<<<(n + 255) / 256, 256, 0, stream>>>(out, n);
}